// MERAFirstOnlyLayer_6949257085058
// MI455X (gfx1250) — compile-verified
//
#include <hip/hip_runtime.h>

typedef __attribute__((ext_vector_type(2))) float v2f;
typedef __attribute__((ext_vector_type(4))) float v4f;
typedef __attribute__((ext_vector_type(8))) float v8f;

// ---------------------------------------------------------------------------
// Kernel 1: contract the 10 small MERA tensors into W (512x16), stored in
// d_ws in WMMA-B "pair" layout: ws[(k>>1)*32 + j*2 + (k&1)] = W[k][j]
// i.e. float2 at pair-index p = (k/2)*16 + j holds (W[2*(k/2)][j], W[2*(k/2)+1][j]).
// ---------------------------------------------------------------------------
__global__ __launch_bounds__(512) void mera_build_w_kernel(
    const float* __restrict__ d0, const float* __restrict__ d1,
    const float* __restrict__ d2, const float* __restrict__ e0,
    const float* __restrict__ e1, const float* __restrict__ e2,
    const float* __restrict__ e3, const float* __restrict__ s0,
    const float* __restrict__ s1, const float* __restrict__ t3,
    float* __restrict__ wsW)
{
    __shared__ float Ms[256];  // Ms[vwxy*16 + ADEF]
    const int tid = threadIdx.x;

    // M[v,w,x,y][A,D,E,F] = sum_{B,C} s0[v,w,A,B] s1[x,y,C,D] t3[B,C,E,F]
    if (tid < 256) {
        const int vwxy = tid >> 4, adef = tid & 15;
        const int v = (vwxy >> 3) & 1, w = (vwxy >> 2) & 1;
        const int xx = (vwxy >> 1) & 1, y = vwxy & 1;
        const int A = (adef >> 3) & 1, D = (adef >> 2) & 1;
        const int E = (adef >> 1) & 1, F = adef & 1;
        float acc = 0.f;
        for (int Bb = 0; Bb < 2; ++Bb)
            for (int Cc = 0; Cc < 2; ++Cc)
                acc += s0[((v * 2 + w) * 2 + A) * 2 + Bb]
                     * s1[((xx * 2 + y) * 2 + Cc) * 2 + D]
                     * t3[((Bb * 2 + Cc) * 2 + E) * 2 + F];
        Ms[tid] = acc;
    }
    __syncthreads();

    // One thread per input index k = (a,c,d,e,f,g,h,m,n)
    const int k = tid;
    const int a = (k >> 8) & 1, c = (k >> 7) & 1, d = (k >> 6) & 1;
    const int e = (k >> 5) & 1, f = (k >> 4) & 1, g = (k >> 3) & 1;
    const int h = (k >> 2) & 1, m = (k >> 1) & 1, n = k & 1;

    // Chain contraction over internal bonds p,q,r,s,t,u for each (v,w,x,y)
    float U[16];
    for (int v = 0; v < 2; ++v) {
        float f1[2];
        for (int q = 0; q < 2; ++q) {
            float s = 0.f;
            for (int p = 0; p < 2; ++p)
                s += e0[(a * 2 + p) * 2 + v]
                   * d0[(((c * 2 + d) * 2 + e) * 2 + p) * 2 + q];
            f1[q] = s;
        }
        for (int w = 0; w < 2; ++w) {
            float f2[2], f3[2];
            for (int rr = 0; rr < 2; ++rr) {
                float s = 0.f;
                for (int q = 0; q < 2; ++q) s += f1[q] * e1[(q * 2 + rr) * 2 + w];
                f2[rr] = s;
            }
            for (int ss = 0; ss < 2; ++ss) {
                float s = 0.f;
                for (int rr = 0; rr < 2; ++rr) s += f2[rr] * d1[((f * 2 + g) * 2 + rr) * 2 + ss];
                f3[ss] = s;
            }
            for (int xx = 0; xx < 2; ++xx) {
                float f4[2], f5[2];
                for (int t = 0; t < 2; ++t) {
                    float s = 0.f;
                    for (int ss = 0; ss < 2; ++ss) s += f3[ss] * e2[(ss * 2 + t) * 2 + xx];
                    f4[t] = s;
                }
                for (int u = 0; u < 2; ++u) {
                    float s = 0.f;
                    for (int t = 0; t < 2; ++t) s += f4[t] * d2[((h * 2 + m) * 2 + t) * 2 + u];
                    f5[u] = s;
                }
                for (int y = 0; y < 2; ++y) {
                    float s = 0.f;
                    for (int u = 0; u < 2; ++u) s += f5[u] * e3[(u * 2 + n) * 2 + y];
                    U[((v * 2 + w) * 2 + xx) * 2 + y] = s;
                }
            }
        }
    }

    // W[k][j] = sum_vwxy U * M ; store pre-swizzled for WMMA B operand
    for (int j = 0; j < 16; ++j) {
        float acc = 0.f;
        for (int vv = 0; vv < 16; ++vv) acc += U[vv] * Ms[vv * 16 + j];
        wsW[(k >> 1) * 32 + j * 2 + (k & 1)] = acc;
    }
}

// ---------------------------------------------------------------------------
// Kernel 2: out[B,16] = X[B,512] * W[512,16] + bias, via V_WMMA_F32_16X16X4_F32.
// 256 threads = 8 wave32 per block; each wave owns a 16x16 output tile.
// X staged through LDS (stride-68 padding -> conflict-free 16x4 A-tile reads).
// Software pipeline: load chunk c+1 to regs while computing on LDS chunk c.
// ---------------------------------------------------------------------------
#define ROWS_PER_BLOCK 128
#define KCHUNK 64
#define LDS_STRIDE 68  // 64 + 4 pad: lane L bank = 4L + const -> all 64 banks distinct

__global__ __launch_bounds__(256) void mera_gemm_kernel(
    const float* __restrict__ X, const float* __restrict__ Wp,
    const float* __restrict__ bias, float* __restrict__ out)
{
    __shared__ float lds_x[ROWS_PER_BLOCK * LDS_STRIDE];  // 34816 B

    const int tid  = threadIdx.x;
    const int lane = tid & 31;
    const int wave = tid >> 5;      // 0..7
    const int n    = lane & 15;     // WMMA column / B lane index
    const int hi   = lane >> 4;     // 0: K=0,1 ; 1: K=2,3 (A/B halves)

    const long long rowblk = (long long)blockIdx.x * ROWS_PER_BLOCK;
    const float* xbase = X + rowblk * 512;

    // Tile-load mapping: 16 consecutive threads read one contiguous 256B row segment
    const int lrow = tid >> 4;        // 0..15
    const int lcol = (tid & 15) * 4;  // 0..60

    v4f r[8];
    // prologue: chunk 0 -> regs -> LDS
    #pragma unroll
    for (int i = 0; i < 8; ++i) {
        const int row = i * 16 + lrow;
        r[i] = *(const v4f*)(xbase + (long long)row * 512 + lcol);
    }
    #pragma unroll
    for (int i = 0; i < 8; ++i) {
        const int row = i * 16 + lrow;
        *(v4f*)&lds_x[row * LDS_STRIDE + lcol] = r[i];
    }
    __syncthreads();

    // Accumulator initialized with bias: lane l, vgpr i -> out row (i + 8*hi), col n
    const float bn = bias[n];
    v8f acc;
    #pragma unroll
    for (int i = 0; i < 8; ++i) acc[i] = bn;

    const int arow = wave * 16 + n;  // this lane's A-matrix row in the tile

    for (int c = 0; c < 8; ++c) {
        if (c < 7) {  // prefetch next K-chunk into registers
            #pragma unroll
            for (int i = 0; i < 8; ++i) {
                const int row = i * 16 + lrow;
                r[i] = *(const v4f*)(xbase + (long long)row * 512 + (c + 1) * KCHUNK + lcol);
            }
        }
        #pragma unroll
        for (int s = 0; s < 16; ++s) {
            const int kk = 4 * s;
            // A (16x4 f32): lane<16 holds A[M=lane][K=0..1], lane>=16 holds A[M][K=2..3]
            const v2f a = *(const v2f*)&lds_x[arow * LDS_STRIDE + kk + 2 * hi];
            // B (4x16 f32) from pre-swizzled W pairs: (W[K0+2hi][n], W[K0+2hi+1][n])
            const v2f b = *(const v2f*)(Wp + (((c * 32 + 2 * s + hi) * 16 + n) << 1));
            acc = __builtin_amdgcn_wmma_f32_16x16x4_f32(
                false, a, false, b, (short)0, acc, false, false);
        }
        __syncthreads();
        if (c < 7) {
            #pragma unroll
            for (int i = 0; i < 8; ++i) {
                const int row = i * 16 + lrow;
                *(v4f*)&lds_x[row * LDS_STRIDE + lcol] = r[i];
            }
            __syncthreads();
        }
    }

    // Store: D vgpr i, lane l -> out[rowblk + wave*16 + i + 8*hi][n]
    float* obase = out + (rowblk + wave * 16 + hi * 8) * 16 + n;
    #pragma unroll
    for (int i = 0; i < 8; ++i) obase[i * 16] = acc[i];
}

extern "C" void kernel_launch(void* const* d_in, const int* in_sizes, int n_in,
                              void* d_out, int out_size, void* d_ws, size_t ws_size,
                              hipStream_t stream) {
    const float* x    = (const float*)d_in[0];
    const float* d0   = (const float*)d_in[1];
    const float* d1   = (const float*)d_in[2];
    const float* d2   = (const float*)d_in[3];
    const float* e0   = (const float*)d_in[4];
    const float* e1   = (const float*)d_in[5];
    const float* e2   = (const float*)d_in[6];
    const float* e3   = (const float*)d_in[7];
    const float* s0   = (const float*)d_in[8];
    const float* s1   = (const float*)d_in[9];
    const float* t3   = (const float*)d_in[10];
    const float* bias = (const float*)d_in[11];
    float* wsW = (float*)d_ws;
    float* out = (float*)d_out;

    const int nb = in_sizes[0] / 512;          // batch rows (131072)
    const int grid = nb / ROWS_PER_BLOCK;      // 1024 blocks

    mera_build_w_kernel<<<dim3(1), dim3(512), 0, stream>>>(
        d0, d1, d2, e0, e1, e2, e3, s0, s1, t3, wsW);
    mera_gemm_kernel<<<dim3(grid), dim3(256), 0, stream>>>(x, wsW, bias, out);
}